// FuzzyIntens_65077344469134
// MI455X (gfx1250) — compile-verified
//
#include <hip/hip_runtime.h>
#include <hip/hip_bf16.h>
#include <stdint.h>

// Problem sizes (fixed by the reference)
#define KB 2048   // batch
#define KD 256    // reduced dim
#define KU 512    // units

// Tiling
#define BM 64     // rows of output tile per workgroup
#define UN 64     // cols of output tile per workgroup
#define DK 32     // D-chunk per pipeline stage
#define NCHUNK (KD / DK)
#define XPITCH (DK + 1)   // sX row pitch in floats (TDM pad: +1 dword per 32-dword row)

typedef unsigned int v4u __attribute__((ext_vector_type(4)));
typedef int          v8i __attribute__((ext_vector_type(8)));
typedef int          v4i __attribute__((ext_vector_type(4)));

// ---------------------------------------------------------------------------
// Build a TDM D# for a 2D tile load (4-byte elements) and issue it.
//   lds_byte_off : byte offset of destination within the workgroup's LDS
//   gptr         : global address of the *tile* start
//   tensor_d0/d1 : tensor extents (elements) for OOB handling
//   stride0      : elements between consecutive dim-1 lines
//   tile_d0/d1   : tile extents (elements); dim0 is the contiguous axis
//   pad_en       : 1 -> insert pad_amount dwords every 2^(pad_interval+1) dwords
//   pad_ivl      : pad_interval code (4 -> every 32 dwords), pad_amount=0 -> 1 dword
// ---------------------------------------------------------------------------
__device__ inline void tdm_load_tile_2d(uint32_t lds_byte_off, const void* gptr,
                                        uint32_t tensor_d0, uint32_t tensor_d1,
                                        uint64_t stride0,
                                        uint32_t tile_d0, uint32_t tile_d1,
                                        uint32_t pad_en, uint32_t pad_ivl) {
  const uint64_t ga = (uint64_t)(uintptr_t)gptr;

  // --- D# group 0 (128b): count=1 | lds_addr | global_addr[56:0] | type=2 ---
  v4u g0;
  g0.x = 1u;                                            // count=1, user desc
  g0.y = lds_byte_off;                                  // lds_addr (bytes)
  g0.z = (uint32_t)ga;                                  // global_addr[31:0]
  g0.w = (uint32_t)((ga >> 32) & 0x01FFFFFFu) | (2u << 30); // addr[56:32], type=2

  // --- D# group 1 (256b) ---
  // q0 bits: [15:0] wg_mask=0 | [17:16] data_size=2 (4B) | [20] pad_en |
  //          [24:22] pad_interval | [31:25] pad_amount=0 |
  //          [47:32] atomic_barrier_addr=0 | [63:48] tensor_dim0[15:0]
  const uint32_t flags = (2u << 16) | (pad_en << 20) | (pad_ivl << 22);
  const uint64_t q0 = (uint64_t)flags | ((uint64_t)(tensor_d0 & 0xFFFFu) << 48);
  // q1 bits: [15:0] tensor_dim0[31:16] | [47:16] tensor_dim1 | [63:48] tile_dim0
  const uint64_t q1 = (uint64_t)(tensor_d0 >> 16)
                    | ((uint64_t)tensor_d1 << 16)
                    | ((uint64_t)(tile_d0 & 0xFFFFu) << 48);
  // q2 bits: [15:0] tile_dim1 | [31:16] tile_dim2=0 | [63:32] stride0[31:0]
  const uint64_t q2 = (uint64_t)(tile_d1 & 0xFFFFu)
                    | ((uint64_t)(uint32_t)stride0 << 32);
  // q3 bits: [15:0] stride0[47:32] | [63:16] tensor_dim1_stride=0 (2D tile)
  const uint64_t q3 = (uint64_t)((stride0 >> 32) & 0xFFFFu);

  v8i g1;
  g1[0] = (int)(uint32_t)q0;  g1[1] = (int)(uint32_t)(q0 >> 32);
  g1[2] = (int)(uint32_t)q1;  g1[3] = (int)(uint32_t)(q1 >> 32);
  g1[4] = (int)(uint32_t)q2;  g1[5] = (int)(uint32_t)(q2 >> 32);
  g1[6] = (int)(uint32_t)q3;  g1[7] = (int)(uint32_t)(q3 >> 32);

  const v4i z4 = (v4i){0, 0, 0, 0};                      // groups 2/3: unused (2D)
  const v8i z8 = (v8i){0, 0, 0, 0, 0, 0, 0, 0};          // extra operand (clang-23 form)

  // amdgpu-toolchain (clang-23) 6-arg form; cpol=0 (default scope/temporal)
  __builtin_amdgcn_tensor_load_to_lds(g0, g1, z4, z4, z8, 0);
}

__device__ __forceinline__ float clamp01(float v) {
  return fminf(fmaxf(v, 0.0f), 1.0f);
}

// ---------------------------------------------------------------------------
// y[b,u] = clip( sum_d exp2( log2(x[b,d]) * exp2(w[d,u]) ) + bias[u], 0, 1 )
// One 256-thread block computes a 64(B) x 64(U) output tile; the D=256
// reduction is streamed in 8 chunks of 32 via TDM double buffering.
// ---------------------------------------------------------------------------
__global__ __launch_bounds__(256)
void fuzzy_intens_kernel(const float* __restrict__ x,
                         const float* __restrict__ w,
                         const float* __restrict__ bias,
                         float* __restrict__ out) {
  // sX: [2][BM][DK] with +1-dword row pad (TDM pad) -> conflict-free column reads
  __shared__ float sX[2][BM * XPITCH];
  __shared__ float sE[2][DK * UN];

  const int un0 = blockIdx.x * UN;
  const int bm0 = blockIdx.y * BM;
  const int tid = (int)threadIdx.x;
  const int tx  = tid & 15;    // 16 threads across U
  const int ty  = tid >> 4;    // 16 threads across B
  const bool issuer = (tid < 32);   // wave 0 drives the TDM

  // ---- stage helpers -------------------------------------------------------
  auto issue_chunk = [&](int k, int buf) {
    // x tile: rows [bm0, bm0+64), cols [k*DK, k*DK+32); contiguous axis = D
    tdm_load_tile_2d((uint32_t)(uintptr_t)&sX[buf][0],
                     x + (size_t)bm0 * KD + (size_t)k * DK,
                     KD, KB, KD, DK, BM,
                     /*pad_en=*/1, /*pad_ivl=*/4);      // +1 dword per 32-dword row
    // w tile: rows [k*DK, k*DK+32), cols [un0, un0+64); contiguous axis = U
    tdm_load_tile_2d((uint32_t)(uintptr_t)&sE[buf][0],
                     w + (size_t)k * DK * KU + un0,
                     KU, KD, KU, UN, DK,
                     /*pad_en=*/0, /*pad_ivl=*/0);
  };

  auto transform_chunk = [&](int buf) {
    // x tile -> log2(x); raw layout matches TDM pad: row pitch = 33 floats
    for (int e = tid; e < BM * DK; e += 256) {
      const int idx = (e >> 5) * XPITCH + (e & (DK - 1));
      sX[buf][idx] = __builtin_amdgcn_logf(sX[buf][idx]);   // v_log_f32 (log2)
    }
    // w tile -> exp2(w)
    for (int e = tid; e < DK * UN; e += 256) {
      sE[buf][e] = __builtin_amdgcn_exp2f(sE[buf][e]);      // v_exp_f32 (exp2)
    }
  };

  float acc[4][4];
#pragma unroll
  for (int i = 0; i < 4; ++i)
#pragma unroll
    for (int j = 0; j < 4; ++j) acc[i][j] = 0.0f;

  auto compute_chunk = [&](int buf) {
#pragma unroll 4
    for (int d = 0; d < DK; ++d) {
      float l[4], e[4];
#pragma unroll
      for (int i = 0; i < 4; ++i) l[i] = sX[buf][(ty * 4 + i) * XPITCH + d];
#pragma unroll
      for (int j = 0; j < 4; ++j) e[j] = sE[buf][d * UN + tx * 4 + j];
#pragma unroll
      for (int i = 0; i < 4; ++i)
#pragma unroll
        for (int j = 0; j < 4; ++j)
          acc[i][j] += __builtin_amdgcn_exp2f(l[i] * e[j]);  // 16 TRANS / 8 LDS reads
    }
  };

  // ---- prologue: chunk 0 ---------------------------------------------------
  if (issuer) {
    issue_chunk(0, 0);
    __builtin_amdgcn_s_wait_tensorcnt(0);
  }
  __syncthreads();
  transform_chunk(0);
  __syncthreads();

  // ---- pipelined main loop: DMA chunk k+1 while computing chunk k ----------
  for (int k = 0; k < NCHUNK; ++k) {
    const int cur = k & 1;
    const int nxt = cur ^ 1;
    if (k + 1 < NCHUNK && issuer) issue_chunk(k + 1, nxt);
    compute_chunk(cur);
    if (k + 1 < NCHUNK) {
      if (issuer) __builtin_amdgcn_s_wait_tensorcnt(0);
      __syncthreads();           // TDM data visible + all reads of nxt done
      transform_chunk(nxt);
      __syncthreads();
    }
  }

  // ---- epilogue: bias + clip, 128-bit stores -------------------------------
  const float4 bv = *(const float4*)(bias + un0 + tx * 4);
  const float bj[4] = {bv.x, bv.y, bv.z, bv.w};
#pragma unroll
  for (int i = 0; i < 4; ++i) {
    float4 o;
    o.x = clamp01(acc[i][0] + bj[0]);
    o.y = clamp01(acc[i][1] + bj[1]);
    o.z = clamp01(acc[i][2] + bj[2]);
    o.w = clamp01(acc[i][3] + bj[3]);
    *(float4*)(out + (size_t)(bm0 + ty * 4 + i) * KU + un0 + tx * 4) = o;
  }
}

extern "C" void kernel_launch(void* const* d_in, const int* in_sizes, int n_in,
                              void* d_out, int out_size, void* d_ws, size_t ws_size,
                              hipStream_t stream) {
  (void)in_sizes; (void)n_in; (void)out_size; (void)d_ws; (void)ws_size;
  const float* x    = (const float*)d_in[0];   // (2048, 256)
  const float* w    = (const float*)d_in[1];   // (256, 512)
  const float* bias = (const float*)d_in[2];   // (512,)
  float*       out  = (float*)d_out;           // (2048, 512)

  dim3 grid(KU / UN, KB / BM);   // (8, 32) = 256 workgroups
  dim3 block(256);
  fuzzy_intens_kernel<<<grid, block, 0, stream>>>(x, w, bias, out);
}